// InternalCoordinateTransform_11390253269584
// MI455X (gfx1250) — compile-verified
//
#include <hip/hip_runtime.h>
#include <stdint.h>
#include <math.h>

// Problem geometry (from the reference): B x (3*2048) fp32, NZ = 2045 rows of z-matrix.
#define N_ATOMS   2048
#define DIMS      (3 * N_ATOMS)      // 6144 floats = 24 KB per row
#define NZ        (N_ATOMS - 3)      // 2045
#define BLOCK     256                // 8 wave32 per WG
#define CHUNKS    ((DIMS / 4) / BLOCK) // 1536 float4 / 256 lanes = 6 per lane
#define PI_F      3.14159265358979323846f

__global__ __launch_bounds__(BLOCK)
void icoord_transform_kernel(const float* __restrict__ x,
                             const int*   __restrict__ zmat,   // NZ x 4: [i4, i4-1, i4-2, i4-3]
                             const float* __restrict__ mean_b,
                             const float* __restrict__ std_b,
                             const float* __restrict__ mean_a,
                             const float* __restrict__ std_a,
                             const float* __restrict__ mean_d,
                             const float* __restrict__ std_d,
                             float* __restrict__ out)
{
    __shared__ float sh[DIMS];                  // one full input row, staged once

    const int b   = blockIdx.x;
    const int tid = threadIdx.x;
    const float* row  = x   + (size_t)b * DIMS; // uniform per block -> lives in SGPR pair
    float*       orow = out + (size_t)b * DIMS;

    // ---- CDNA5 async copy: global -> LDS, 128b per lane, 6 issues per lane ----
    // Low 32 bits of a flat pointer into the LDS aperture are the LDS byte address.
    unsigned ldsBase = (unsigned)(uintptr_t)(&sh[0]);
#pragma unroll
    for (int j = 0; j < CHUNKS; ++j) {
        unsigned byteOff = (unsigned)(tid + j * BLOCK) * 16u;  // coalesced b128 stream
        unsigned ldsAddr = ldsBase + byteOff;
        asm volatile("global_load_async_to_lds_b128 %0, %1, %2"
                     :
                     : "v"(ldsAddr), "v"(byteOff), "s"(row)
                     : "memory");
    }
    asm volatile("s_wait_asynccnt 0" ::: "memory");
    __syncthreads();

    // Pass-through columns 0..8 (the only columns the z-matrix never writes).
    if (tid < 9) orow[tid] = sh[tid];

    // ---- per-internal-coordinate math (each point reused 4x from LDS) ----
    const int4* zm = (const int4*)zmat;
    for (int i = tid; i < NZ; i += BLOCK) {
        int4 z = zm[i];                 // x=i4, y=i1, z=i2, w=i3
        const int i4 = z.x, i1 = z.y, i2 = z.z, i3 = z.w;

        float p1x = sh[3*i1], p1y = sh[3*i1+1], p1z = sh[3*i1+2];
        float p2x = sh[3*i2], p2y = sh[3*i2+1], p2z = sh[3*i2+2];
        float p3x = sh[3*i3], p3y = sh[3*i3+1], p3z = sh[3*i3+2];
        float p4x = sh[3*i4], p4y = sh[3*i4+1], p4z = sh[3*i4+2];

        // bond = |p1 - p4|
        float dx = p1x - p4x, dy = p1y - p4y, dz = p1z - p4z;
        float bond = sqrtf(dx*dx + dy*dy + dz*dz);

        // angle = acos( dot(normalize(p2-p1), normalize(p4-p1)) )
        float ux = p2x - p1x, uy = p2y - p1y, uz = p2z - p1z;
        float vx = p4x - p1x, vy = p4y - p1y, vz = p4z - p1z;   // b2 as well
        float inu = 1.0f / sqrtf(ux*ux + uy*uy + uz*uz);
        float inv = 1.0f / sqrtf(vx*vx + vy*vy + vz*vz);
        float cosang = (ux*vx + uy*vy + uz*vz) * inu * inv;
        float angle = acosf(cosang);

        // dihedral
        float b0x = p3x - p2x, b0y = p3y - p2y, b0z = p3z - p2z;
        float w1x = p1x - p2x, w1y = p1y - p2y, w1z = p1z - p2z;
        float in1 = 1.0f / sqrtf(w1x*w1x + w1y*w1y + w1z*w1z);
        float b1x = w1x * in1, b1y = w1y * in1, b1z = w1z * in1;

        float c0 = b0x*b1x + b0y*b1y + b0z*b1z;
        float vvx = b0x - c0*b1x, vvy = b0y - c0*b1y, vvz = b0z - c0*b1z;
        float c1 = vx*b1x + vy*b1y + vz*b1z;               // dot(b2, b1)
        float wwx = vx - c1*b1x, wwy = vy - c1*b1y, wwz = vz - c1*b1z;

        float xc = vvx*wwx + vvy*wwy + vvz*wwz;
        float cx = b1y*vvz - b1z*vvy;
        float cy = b1z*vvx - b1x*vvz;
        float cz = b1x*vvy - b1y*vvx;
        float yc = cx*wwx + cy*wwy + cz*wwz;
        float dih = -atan2f(yc, xc);

        // normalization (tables are L2-resident; coalesced per-lane loads)
        float bn = (bond  - mean_b[i]) / std_b[i];
        float an = (angle - mean_a[i]) / std_a[i];
        float d  = dih - mean_d[i];
        if (d < -PI_F) d += 2.0f * PI_F;
        if (d >  PI_F) d -= 2.0f * PI_F;
        float dn = d / std_d[i];

        // contiguous 12B per lane -> wave writes one contiguous 384B span
        const int c = 3 * i4;
        orow[c]     = bn;
        orow[c + 1] = an;
        orow[c + 2] = dn;
    }
}

extern "C" void kernel_launch(void* const* d_in, const int* in_sizes, int n_in,
                              void* d_out, int out_size, void* d_ws, size_t ws_size,
                              hipStream_t stream)
{
    const float* x      = (const float*)d_in[0];
    const int*   zmat   = (const int*)  d_in[1];
    const float* mean_b = (const float*)d_in[2];
    const float* std_b  = (const float*)d_in[3];
    const float* mean_a = (const float*)d_in[4];
    const float* std_a  = (const float*)d_in[5];
    const float* mean_d = (const float*)d_in[6];
    const float* std_d  = (const float*)d_in[7];
    float*       out    = (float*)d_out;

    const int B = in_sizes[0] / DIMS;   // 4096
    icoord_transform_kernel<<<B, BLOCK, 0, stream>>>(
        x, zmat, mean_b, std_b, mean_a, std_a, mean_d, std_d, out);
}